// G4_67061619360101
// MI455X (gfx1250) — compile-verified
//
#include <hip/hip_runtime.h>

// G4 angular symmetry function, factorized into WMMA GEMMs.
//
// out[b,i] = sum_k ( (W@W)[i,k] + sum_d (U_d@U_d)[i,k] ) * W[i,k]
//   W_ij   = exp(-eta*d_ij^2) * tanh(1 - d_ij/R)^3 * [d_ij <= R]
//   U_d    = v[:,:,d] * W / d
//
// B=16 batches -> 16 workgroups; each workgroup stages its four 96x96 f32
// matrices in LDS (~148 KB of the WGP's 320 KB) and runs 4 chained
// 96x96x96 GEMMs with V_WMMA_F32_16X16X4_F32.

typedef float v2f __attribute__((ext_vector_type(2)));
typedef float v8f __attribute__((ext_vector_type(8)));

#define N       96
#define BATCH   16
#define NT      6        // 96 / 16 column tiles per row strip
#define THREADS 192      // 6 waves * 32 lanes (wave32); one wave per row strip
#define CUTOFF  5.0f
#define ETA     0.5f
// zeta = 1, lambda = 1  ->  coefficient 2^(1-zeta) = 1, cosine_term = 1 + cos

__launch_bounds__(THREADS)
__global__ void g4_angular_wmma(const float* __restrict__ dvec,   // (B,N,N,3)
                                const float* __restrict__ dist,   // (B,N,N)
                                float* __restrict__ out) {        // (B,N)
  __shared__ float sW[N * N];        // 36 KB  radial*exp weight matrix
  __shared__ float sU[3][N * N];     // 108 KB weighted unit direction matrices
  __shared__ float sSum[N];          // per-row reduction accumulators

  const int b   = blockIdx.x;
  const int tid = threadIdx.x;

  if (tid < N) sSum[tid] = 0.0f;

  const float* distB = dist + (size_t)b * N * N;
  const float* dvecB = dvec + (size_t)b * N * N * 3;

  // ---- Stage 0: elementwise transform of inputs into LDS -------------------
  for (int idx = tid; idx < N * N; idx += THREADS) {
    float d = distB[idx];
    float e = __expf(-ETA * d * d);
    float t = tanhf(1.0f - d * (1.0f / CUTOFF));
    float r = t * t * t;
    r = (d <= CUTOFF) ? r : 0.0f;
    float w = e * r;
    sW[idx] = w;
    float inv = (d > 0.0f) ? (w / d) : 0.0f;   // guard matches nan_to_num intent
    sU[0][idx] = dvecB[idx * 3 + 0] * inv;
    sU[1][idx] = dvecB[idx * 3 + 1] * inv;
    sU[2][idx] = dvecB[idx * 3 + 2] * inv;
  }
  __syncthreads();

  // ---- Stage 1: S = W@W + U0@U0 + U1@U1 + U2@U2 via f32 WMMA ---------------
  const int lane = tid & 31;
  const int m    = lane & 15;        // row within 16-row tile (A), col (B/C/D)
  const int half = lane >> 4;        // lane half selects K pair / upper M rows
  const int i0   = (tid >> 5) * 16;  // this wave's output row strip

  v8f acc[NT] = {};                  // 6 tiles of 16x16 f32 accumulators

#pragma unroll
  for (int mat = 0; mat < 4; ++mat) {
    const float* X = (mat == 0) ? sW : sU[mat - 1];
    for (int k0 = 0; k0 < N; k0 += 4) {
      // A fragment 16x4: lanes 0-15 hold K={k0,k0+1}, lanes 16-31 K={k0+2,k0+3}
      v2f a;
      a.x = X[(i0 + m) * N + (k0 + 2 * half + 0)];
      a.y = X[(i0 + m) * N + (k0 + 2 * half + 1)];
#pragma unroll
      for (int t = 0; t < NT; ++t) {
        // B fragment 4x16: VGPR v holds row K = k0 + 2*half + v, N = lane&15
        v2f bf;
        bf.x = X[(k0 + 2 * half + 0) * N + t * 16 + m];
        bf.y = X[(k0 + 2 * half + 1) * N + t * 16 + m];
        acc[t] = __builtin_amdgcn_wmma_f32_16x16x4_f32(
            /*neg_a=*/false, a, /*neg_b=*/false, bf,
            /*c_mod=*/(short)0, acc[t],
            /*reuse_a=*/false, /*reuse_b=*/false);
      }
    }
  }

  // ---- Stage 2: multiply by W elementwise and reduce rows ------------------
  // C/D layout: VGPR r holds M = r (lanes 0-15) or r+8 (lanes 16-31), N = lane&15
#pragma unroll
  for (int t = 0; t < NT; ++t) {
#pragma unroll
    for (int r = 0; r < 8; ++r) {
      int row = i0 + r + 8 * half;
      int col = t * 16 + m;
      float v = acc[t][r] * sW[row * N + col];
      atomicAdd(&sSum[row], v);      // ds_add_f32
    }
  }
  __syncthreads();

  if (tid < N) out[(size_t)b * N + tid] = sSum[tid];   // coefficient = 1
}

extern "C" void kernel_launch(void* const* d_in, const int* in_sizes, int n_in,
                              void* d_out, int out_size, void* d_ws, size_t ws_size,
                              hipStream_t stream) {
  const float* dvec = (const float*)d_in[0];   // distance_vectors (B,N,N,3)
  const float* dist = (const float*)d_in[1];   // distances        (B,N,N)
  float* out = (float*)d_out;                  // (B,N)
  (void)in_sizes; (void)n_in; (void)out_size; (void)d_ws; (void)ws_size;
  g4_angular_wmma<<<BATCH, THREADS, 0, stream>>>(dvec, dist, out);
}